// KernelAttentionProduct_16965120819261
// MI455X (gfx1250) — compile-verified
//
#include <hip/hip_runtime.h>

typedef __attribute__((ext_vector_type(2))) float v2f;
typedef __attribute__((ext_vector_type(8))) float v8f;
typedef __attribute__((ext_vector_type(4))) int   v4i;

constexpr int Sc = 4096;   // sequence length
constexpr int Dc = 64;     // head dim
constexpr int BH = 64;     // B*H = 4*16
constexpr int SPLIT = 8;   // S-splits for the kv reduction
constexpr float EPS = 1e-5f;

// gfx1250 async global->LDS path (ASYNCcnt-tracked direct-to-LDS copies).
// Signature (from hipcc diagnostic): (v4i AS1*, v4i AS3*, imm offset, imm cpol)
#if defined(__gfx1250__) && __has_builtin(__builtin_amdgcn_global_load_async_to_lds_b128) && __has_builtin(__builtin_amdgcn_s_wait_asynccnt)
#define USE_ASYNC_LDS 1
#define AS1 __attribute__((address_space(1)))
#define AS3 __attribute__((address_space(3)))
#define ASYNC_CP_B128(gsrc, ldst) \
    __builtin_amdgcn_global_load_async_to_lds_b128((AS1 v4i*)(gsrc), (AS3 v4i*)(ldst), 0, 0)
#else
#define USE_ASYNC_LDS 0
#endif

// ---------------------------------------------------------------------------
// Kernel 0: zero workspace (kv^T accumulators + ksum accumulators)
// ---------------------------------------------------------------------------
__global__ void zero_ws_kernel(float* __restrict__ p, int n) {
    int i = blockIdx.x * blockDim.x + threadIdx.x;
    if (i < n) p[i] = 0.0f;
}

// ---------------------------------------------------------------------------
// Kernel 1: kv^T[bh][e][d] = sum_s V[bh,s,e] * K[bh,s,d]   (A = V^T, B = K)
//           ksum[bh][d]    = sum_s K[bh,s,d]
// grid = (SPLIT, BH), block = 256 (8 waves). Each wave owns 2 of 16 tiles.
// ---------------------------------------------------------------------------
__global__ __launch_bounds__(256) void kv_reduce_kernel(
    const float* __restrict__ K, const float* __restrict__ V,
    float* __restrict__ kvT, float* __restrict__ ksum)
{
    __shared__ float ldsK[16 * Dc];
    __shared__ float ldsV[16 * Dc];
    __shared__ float ldsRed[256];

    const int bh    = blockIdx.y;
    const int chunk = blockIdx.x;
    const int tid   = threadIdx.x;
    const int lane  = tid & 31;
    const int w     = tid >> 5;
    const int m     = lane & 15;       // row/col within tile
    const int h2    = lane >> 4;       // half-wave select (k-base = 2*h2)

    const int iM = w >> 1;             // e-tile index (A-matrix rows, from V)
    const int j0 = (w & 1) * 2;        // first d-tile index (B-matrix cols, from K)

    const float* Kb = K + (size_t)bh * Sc * Dc;
    const float* Vb = V + (size_t)bh * Sc * Dc;
    const int s_begin = chunk * (Sc / SPLIT);   // 512 rows per block

    v8f acc0 = {};
    v8f acc1 = {};

    float ks_acc = 0.0f;
    const int d_col = tid & 63;
    const int r_off = tid >> 6;

    // staging: each thread moves one float4 of K and V per stage
    const int li = tid * 4;     // element offset within 16x64 stage
    const int lr = li >> 6;     // stage row
    const int lc = li & 63;     // stage col

    for (int stage = 0; stage < (Sc / SPLIT) / 16; ++stage) {
        const int sr = s_begin + stage * 16;
        __syncthreads();   // previous stage fully consumed
#if USE_ASYNC_LDS
        // Direct global->LDS async copy (no VGPR round-trip, ASYNCcnt tracked)
        ASYNC_CP_B128(Kb + (size_t)(sr + lr) * Dc + lc, ldsK + li);
        ASYNC_CP_B128(Vb + (size_t)(sr + lr) * Dc + lc, ldsV + li);
        __builtin_amdgcn_s_wait_asynccnt(0);
#else
        float4 kq = *(const float4*)(Kb + (size_t)(sr + lr) * Dc + lc);
        float4 vq = *(const float4*)(Vb + (size_t)(sr + lr) * Dc + lc);
        *(float4*)(ldsK + li) = kq;
        *(float4*)(ldsV + li) = vq;
        if (stage + 1 < (Sc / SPLIT) / 16) {
            __builtin_prefetch(Kb + (size_t)(sr + 16 + lr) * Dc + lc, 0, 1);
            __builtin_prefetch(Vb + (size_t)(sr + 16 + lr) * Dc + lc, 0, 1);
        }
#endif
        __syncthreads();

        // ksum partial: 4 threads per column, 4 rows each
        #pragma unroll
        for (int r = 0; r < 4; ++r)
            ks_acc += ldsK[(r_off + 4 * r) * Dc + d_col];

        // 4 k-steps of depth 4 over the 16 staged rows
        #pragma unroll
        for (int kk = 0; kk < 4; ++kk) {
            const int r0 = kk * 4 + 2 * h2;
            v2f a, b0, b1;
            a.x  = ldsV[(r0    ) * Dc + 16 * iM + m];
            a.y  = ldsV[(r0 + 1) * Dc + 16 * iM + m];
            b0.x = ldsK[(r0    ) * Dc + 16 * j0 + m];
            b0.y = ldsK[(r0 + 1) * Dc + 16 * j0 + m];
            b1.x = ldsK[(r0    ) * Dc + 16 * (j0 + 1) + m];
            b1.y = ldsK[(r0 + 1) * Dc + 16 * (j0 + 1) + m];
            acc0 = __builtin_amdgcn_wmma_f32_16x16x4_f32(false, a, false, b0,
                                                         (short)0, acc0, false, false);
            acc1 = __builtin_amdgcn_wmma_f32_16x16x4_f32(false, a, false, b1,
                                                         (short)0, acc1, false, false);
        }
    }

    // reduce ksum partials across the block, one atomic per column
    ldsRed[tid] = ks_acc;
    __syncthreads();
    if (tid < 64) {
        float s = ldsRed[tid] + ldsRed[tid + 64] + ldsRed[tid + 128] + ldsRed[tid + 192];
        atomicAdd(&ksum[bh * Dc + tid], s);
    }

    // accumulate this split's partial kv^T tiles into workspace
    float* kvb = kvT + (size_t)bh * Dc * Dc;
    #pragma unroll
    for (int v = 0; v < 8; ++v) {
        const int e = 16 * iM + v + 8 * h2;   // C-layout: VGPR v, half-wave h2
        atomicAdd(&kvb[e * Dc + 16 * j0 + m],       acc0[v]);
        atomicAdd(&kvb[e * Dc + 16 * (j0 + 1) + m], acc1[v]);
    }
}

// ---------------------------------------------------------------------------
// Kernel 2: out[s,e] = (Q[s,:]/(Q[s,:].ksum + eps)) @ kv
// grid = (S/128, BH), block = 256 (8 waves x 16 rows each)
// ---------------------------------------------------------------------------
__global__ __launch_bounds__(256) void context_kernel(
    const float* __restrict__ Q, const float* __restrict__ kvT,
    const float* __restrict__ ksum, float* __restrict__ Out)
{
    __shared__ float lds_kv[Dc * Dc];        // kv^T, [e][d]
    __shared__ float lds_ks[Dc];
    __shared__ float lds_q[8 * 16 * Dc];     // per-wave staged/normalized Q rows

    const int bh   = blockIdx.y;
    const int tid  = threadIdx.x;
    const int lane = tid & 31;
    const int w    = tid >> 5;
    const int m    = lane & 15;
    const int h2   = lane >> 4;

    const int s0 = blockIdx.x * 128 + w * 16;
    const float* Qb = Q + (size_t)bh * Sc * Dc + (size_t)s0 * Dc;
    float* qs = lds_q + w * (16 * Dc);

    // cache kv^T (16 KB) + this wave's 16 Q rows in LDS
    {
        const float* src = kvT + (size_t)bh * Dc * Dc;
#if USE_ASYNC_LDS
        #pragma unroll
        for (int it = 0; it < 4; ++it) {
            int off = it * 1024 + tid * 4;
            ASYNC_CP_B128(src + off, lds_kv + off);
        }
        #pragma unroll
        for (int it = 0; it < 8; ++it) {
            int off = it * 128 + lane * 4;
            ASYNC_CP_B128(Qb + off, qs + off);
        }
#else
        #pragma unroll
        for (int it = 0; it < 4; ++it) {
            int off = it * 1024 + tid * 4;
            *(float4*)(lds_kv + off) = *(const float4*)(src + off);
        }
        #pragma unroll
        for (int it = 0; it < 8; ++it) {
            int off = it * 128 + lane * 4;
            *(float4*)(qs + off) = *(const float4*)(Qb + off);
        }
#endif
        if (tid < Dc) lds_ks[tid] = ksum[bh * Dc + tid];
#if USE_ASYNC_LDS
        __builtin_amdgcn_s_wait_asynccnt(0);
#endif
    }
    __syncthreads();

    // denominator: 2 lanes per row, half-row dot each, combined via wave32 shuffle
    float part = 0.0f;
    #pragma unroll
    for (int c = 0; c < 32; ++c)
        part += qs[m * Dc + h2 * 32 + c] * lds_ks[h2 * 32 + c];
    float other = __shfl_xor(part, 16, 32);
    float rinv = 1.0f / (part + other + EPS);
    #pragma unroll
    for (int c = 0; c < 32; ++c)
        qs[m * Dc + h2 * 32 + c] *= rinv;
    __syncthreads();

    // out tile row-block (16 x 64) = qhat (16 x 64) @ kv (64 x 64)
    v8f acc[4] = {{}, {}, {}, {}};
    #pragma unroll 4
    for (int kk = 0; kk < 16; ++kk) {
        // A fragment: lane holds q[row m][k = 4*kk + 2*h2 + {0,1}]
        v2f a = *(const v2f*)(qs + m * Dc + kk * 4 + 2 * h2);
        #pragma unroll
        for (int j = 0; j < 4; ++j) {
            // B fragment: B[k][n] = kv[d=k][e=16j+n] = kvT[16j+n][k] (contiguous pair)
            v2f b = *(const v2f*)(lds_kv + (16 * j + m) * Dc + kk * 4 + 2 * h2);
            acc[j] = __builtin_amdgcn_wmma_f32_16x16x4_f32(false, a, false, b,
                                                           (short)0, acc[j], false, false);
        }
    }

    float* Ob = Out + (size_t)bh * Sc * Dc + (size_t)s0 * Dc;
    #pragma unroll
    for (int v = 0; v < 8; ++v) {
        const int row = v + 8 * h2;
        #pragma unroll
        for (int j = 0; j < 4; ++j)
            Ob[row * Dc + 16 * j + m] = acc[j][v];
    }
}

// ---------------------------------------------------------------------------
extern "C" void kernel_launch(void* const* d_in, const int* in_sizes, int n_in,
                              void* d_out, int out_size, void* d_ws, size_t ws_size,
                              hipStream_t stream) {
    const float* Q = (const float*)d_in[0];
    const float* K = (const float*)d_in[1];
    const float* V = (const float*)d_in[2];
    float* Out = (float*)d_out;

    float* ws   = (float*)d_ws;
    float* kvT  = ws;                       // BH * 64 * 64 floats
    float* ksum = ws + BH * Dc * Dc;        // BH * 64 floats

    const int nzero = BH * Dc * Dc + BH * Dc;
    zero_ws_kernel<<<(nzero + 255) / 256, 256, 0, stream>>>(ws, nzero);
    kv_reduce_kernel<<<dim3(SPLIT, BH), 256, 0, stream>>>(K, V, kvT, ksum);
    context_kernel<<<dim3(Sc / 128, BH), 256, 0, stream>>>(Q, kvT, ksum, Out);
}